// BaseGMVAE_420906795094
// MI455X (gfx1250) — compile-verified
//
#include <hip/hip_runtime.h>
#include <hip/hip_bf16.h>
#include <math.h>

// ---------------------------------------------------------------------------
// GMVAE posterior:  logits[b,k] = -0.5*sum_d((q-mu)^2/var + lv + log2pi) + log(1/K)
// Reformulated as one fp32 GEMM  [B,2D] x [2D,16]  via V_WMMA_F32_16X16X4_F32:
//   A'[b, d    ] = q_z[b,d]        W[d    ,k] =  mu[k,d] * e^{-lv}
//   A'[b, D + d] = q_z[b,d]^2      W[D + d,k] = -0.5     * e^{-lv}
// plus a per-k constant.  Softmax + argmax done in-fragment with shfl_xor.
// ---------------------------------------------------------------------------

#define B_ROWS   65536
#define D_DIM    256
#define K_MIX    10
#define K_PAD    16
#define LOG_2PI  1.8378770664093453f
#define LOG_INVK (-2.302585092994046f)   // log(1/10)

typedef __attribute__((ext_vector_type(2))) float v2f;
typedef __attribute__((ext_vector_type(8))) float v8f;

// W fragment-row permutation: WMMA step t (0..63), slot j (0..3) maps to column
//   col(t,j) = 8*(t>>1) + 4*(j>>1) + 2*(t&1) + (j&1)
// so that a lane's global_load_b128 of 4 consecutive floats feeds 2 WMMA steps.
__device__ __forceinline__ int wslot_of_col(int d) {
    int kk8 = d >> 3, w = d & 7;
    int halfsel = w >> 2, rem = w & 3;
    int r = rem >> 1, jb = rem & 1;
    return ((2 * kk8 + r) << 2) + 2 * halfsel + jb;   // t*4 + j
}

// ---- prep: build permuted W1/W2 [256 slots][16 cols] + per-k constant -------
__global__ __launch_bounds__(256)
void gmvae_prep_kernel(const float* __restrict__ mu_table,
                       const float* __restrict__ logvar_table,
                       float* __restrict__ w1,      // [256*16]
                       float* __restrict__ w2,      // [256*16]
                       float* __restrict__ constk)  // [16]
{
    const int k = blockIdx.x;      // 0..15 (10..15 are zero padding)
    const int d = threadIdx.x;     // 0..255
    __shared__ float red[256];

    const int slot = wslot_of_col(d);
    float partial = 0.0f;
    if (k < K_MIX) {
        const float m  = mu_table[k * D_DIM + d];
        const float lv = logvar_table[k * D_DIM + d];
        const float iv = __expf(-lv);
        w1[slot * K_PAD + k] = m * iv;
        w2[slot * K_PAD + k] = -0.5f * iv;
        partial = m * m * iv + lv + LOG_2PI;
    } else {
        w1[slot * K_PAD + k] = 0.0f;
        w2[slot * K_PAD + k] = 0.0f;
    }
    red[d] = partial;
    __syncthreads();
    #pragma unroll
    for (int s = 128; s > 0; s >>= 1) {
        if (d < s) red[d] += red[d + s];
        __syncthreads();
    }
    if (d == 0)
        constk[k] = (k < K_MIX) ? (-0.5f * red[0] + LOG_INVK) : 0.0f;
}

// ---- main: 8 waves/block, 16 batch rows per wave ---------------------------
__global__ __launch_bounds__(256)
void gmvae_wmma_kernel(const float* __restrict__ q_z,
                       const float* __restrict__ w1g,
                       const float* __restrict__ w2g,
                       const float* __restrict__ constg,
                       float* __restrict__ out_logit,  // [B*10]
                       float* __restrict__ out_qy,     // [B*10]
                       float* __restrict__ out_ind)    // [B]
{
    __shared__ float sW1[256 * K_PAD];
    __shared__ float sW2[256 * K_PAD];
    __shared__ float sC[K_PAD];

    const int tid = threadIdx.x;
    for (int i = tid; i < 256 * K_PAD; i += 256) {
        sW1[i] = w1g[i];
        sW2[i] = w2g[i];
    }
    if (tid < K_PAD) sC[tid] = constg[tid];
    __syncthreads();

    const int wave = tid >> 5;
    const int lane = tid & 31;
    const int half = lane >> 4;          // selects K-slots {0,1} vs {2,3}
    const int n    = lane & 15;          // mixture-component column
    const int row_base = blockIdx.x * 128 + wave * 16;

    // A fragment source: lane covers row (row_base + n), 16B chunk per 8 cols
    const float* aptr = q_z + (size_t)(row_base + n) * D_DIM + 4 * half;

    v8f acc = {};
    #pragma unroll 4
    for (int kk8 = 0; kk8 < 32; ++kk8) {
        const float4 a = *(const float4*)(aptr + kk8 * 8);
        const int t0 = kk8 * 2;
        #pragma unroll
        for (int r = 0; r < 2; ++r) {
            v2f av;
            av.x = (r == 0) ? a.x : a.z;
            av.y = (r == 0) ? a.y : a.w;
            v2f av2 = av * av;
            const int rowi = ((t0 + r) << 2) + 2 * half;  // fragment row in W
            v2f b1, b2;
            b1.x = sW1[rowi * K_PAD + n];
            b1.y = sW1[(rowi + 1) * K_PAD + n];
            b2.x = sW2[rowi * K_PAD + n];
            b2.y = sW2[(rowi + 1) * K_PAD + n];
            acc = __builtin_amdgcn_wmma_f32_16x16x4_f32(
                false, av,  false, b1, (short)0, acc, false, false);
            acc = __builtin_amdgcn_wmma_f32_16x16x4_f32(
                false, av2, false, b2, (short)0, acc, false, false);
        }
    }

    // Epilogue: C/D layout -> lane half h, VGPR v holds (row_base + v + 8h, n)
    const float cadd = sC[n];
    #pragma unroll
    for (int v = 0; v < 8; ++v) {
        const int row = row_base + v + half * 8;
        const float logit = acc[v] + cadd;
        if (n < K_MIX) out_logit[row * K_MIX + n] = logit;

        const float x = (n < K_MIX) ? logit : -INFINITY;
        // row max over 16-lane group
        float mx = x;
        #pragma unroll
        for (int m = 1; m < 16; m <<= 1) mx = fmaxf(mx, __shfl_xor(mx, m));
        const float e = __expf(x - mx);   // 0 for padded lanes
        float s = e;
        #pragma unroll
        for (int m = 1; m < 16; m <<= 1) s += __shfl_xor(s, m);
        if (n < K_MIX) out_qy[row * K_MIX + n] = e / s;

        // first-occurrence argmax
        float bv = x;
        int   bi = n;
        #pragma unroll
        for (int m = 1; m < 16; m <<= 1) {
            const float ov = __shfl_xor(bv, m);
            const int   oi = __shfl_xor(bi, m);
            if (ov > bv || (ov == bv && oi < bi)) { bv = ov; bi = oi; }
        }
        if (n == 0) out_ind[row] = (float)bi;
    }
}

extern "C" void kernel_launch(void* const* d_in, const int* in_sizes, int n_in,
                              void* d_out, int out_size, void* d_ws, size_t ws_size,
                              hipStream_t stream) {
    const float* q_z    = (const float*)d_in[0];
    const float* mu     = (const float*)d_in[1];
    const float* logvar = (const float*)d_in[2];

    float* w1     = (float*)d_ws;                    // 256*16 floats
    float* w2     = w1 + 256 * K_PAD;                // 256*16 floats
    float* constk = w2 + 256 * K_PAD;                // 16 floats

    float* out       = (float*)d_out;
    float* out_logit = out;                          // [B*K]
    float* out_qy    = out + B_ROWS * K_MIX;         // [B*K]
    float* out_ind   = out + 2 * B_ROWS * K_MIX;     // [B]

    gmvae_prep_kernel<<<K_PAD, 256, 0, stream>>>(mu, logvar, w1, w2, constk);
    gmvae_wmma_kernel<<<B_ROWS / 128, 256, 0, stream>>>(
        q_z, w1, w2, constk, out_logit, out_qy, out_ind);
}